// Hybrid_SOTA_SNN_14242111553595
// MI455X (gfx1250) — compile-verified
//
#include <hip/hip_runtime.h>
#include <hip/hip_bf16.h>

// ---------------- problem constants (from reference) ----------------
#define BB 128
#define NN 64
#define TT_LEN 1000
#define HH 16
// TAU=2 -> leak = 0.5 ; VTH = 0.5 ; H^-0.5 = 0.25

typedef __attribute__((ext_vector_type(2))) float v2f;
typedef __attribute__((ext_vector_type(8))) float v8f;

// ---------------- kernel A: energy[b,n] = var(x[b,n,:], ddof=1) ----------------
__global__ void k_energy(const float* __restrict__ x, float* __restrict__ energy) {
    int tid  = threadIdx.x;
    int row  = blockIdx.x * 8 + (tid >> 5);   // (b*N+n)
    int lane = tid & 31;
    const float* p = x + (size_t)row * TT_LEN;
    float s = 0.f, q = 0.f;
    for (int t = lane; t < TT_LEN; t += 32) { float v = p[t]; s += v; q += v * v; }
    #pragma unroll
    for (int off = 16; off; off >>= 1) {
        s += __shfl_xor(s, off, 32);
        q += __shfl_xor(q, off, 32);
    }
    if (lane == 0)
        energy[row] = (q - s * s * (1.0f / TT_LEN)) * (1.0f / (TT_LEN - 1));
}

// ---------------- kernel B: attn + diag mask + softmax -> w (B,N,N) ----------------
// attn[b,i,j] = 0.25*(c1*e_i*e_j + c2*e_i + c3*e_j + c4) with c* = dots over H
__global__ void k_attn(const float* __restrict__ energy,
                       const float* __restrict__ qw, const float* __restrict__ qb,
                       const float* __restrict__ kw, const float* __restrict__ kb,
                       float* __restrict__ wout) {
    __shared__ float e[NN];
    int b = blockIdx.x, i = threadIdx.x;
    float c1 = 0.f, c2 = 0.f, c3 = 0.f, c4 = 0.f;
    #pragma unroll
    for (int h = 0; h < HH; ++h) {
        float q = qw[h], Qb = qb[h], k = kw[h], Kb = kb[h];
        c1 += q * k; c2 += q * Kb; c3 += Qb * k; c4 += Qb * Kb;
    }
    e[i] = energy[b * NN + i];
    __syncthreads();
    float ei = e[i];
    const float sc = 0.25f;
    float m = -1e30f;
    for (int j = 0; j < NN; ++j) {
        float a = (j == i) ? -1e9f : sc * (c1 * ei * e[j] + c2 * ei + c3 * e[j] + c4);
        m = fmaxf(m, a);
    }
    float ssum = 0.f;
    for (int j = 0; j < NN; ++j) {
        float a = (j == i) ? -1e9f : sc * (c1 * ei * e[j] + c2 * ei + c3 * e[j] + c4);
        ssum += __expf(a - m);
    }
    float inv = 1.0f / ssum;
    float* wr = wout + (size_t)b * NN * NN + (size_t)i * NN;
    for (int j = 0; j < NN; ++j) {
        float a = (j == i) ? -1e9f : sc * (c1 * ei * e[j] + c2 * ei + c3 * e[j] + c4);
        wr[j] = __expf(a - m) * inv;
    }
}

// ---------------- kernel C: z_gcn = w @ xs + x  via V_WMMA_F32_16X16X4_F32 ----------------
// One wave per 16(n) x 16(t) tile, K=64 in steps of 4.
__global__ void k_gcn(const float* __restrict__ wmat, const float* __restrict__ x,
                      float* __restrict__ zg) {
    const int NTILE = NN / 16;               // 4
    const int TTILE = (TT_LEN + 15) / 16;    // 63
    int tid  = threadIdx.x;
    int lane = tid & 31;
    int tile = blockIdx.x * (blockDim.x >> 5) + (tid >> 5);
    int b    = tile / (NTILE * TTILE);
    int rem  = tile % (NTILE * TTILE);
    int n0   = (rem / TTILE) * 16;
    int t0   = (rem % TTILE) * 16;
    int half = lane >> 4;        // 0: K pair (0,1) / M rows r..r ; 1: K pair (2,3) / M rows r+8
    int l15  = lane & 15;

    const float* wb = wmat + (size_t)b * NN * NN;
    const float* xb = x    + (size_t)b * NN * TT_LEN;

    int  t   = t0 + l15;
    bool tin = t < TT_LEN;
    int  tc  = tin ? t : (TT_LEN - 1);

    v8f acc = {};
    #pragma unroll
    for (int j0 = 0; j0 < NN; j0 += 4) {
        int ka = j0 + half * 2;
        // A 16x4 f32 layout: lanes0-15 VGPR0/1 = K0/K1 at M=lane; lanes16-31 = K2/K3
        v2f av;
        av.x = wb[(n0 + l15) * NN + ka];
        av.y = wb[(n0 + l15) * NN + ka + 1];
        // B 4x16 f32 layout: VGPR0 lanes0-15 K=0 / lanes16-31 K=2; VGPR1: K=1 / K=3
        v2f bv;
        float b0 = xb[(size_t)ka * TT_LEN + tc];
        float b1 = xb[(size_t)(ka + 1) * TT_LEN + tc];
        bv.x = tin ? b0 : 0.f;
        bv.y = tin ? b1 : 0.f;
        acc = __builtin_amdgcn_wmma_f32_16x16x4_f32(false, av, false, bv,
                                                    (short)0, acc, false, false);
    }
    if (tin) {
        #pragma unroll
        for (int r = 0; r < 8; ++r) {
            int n = n0 + r + half * 8;               // C/D layout
            size_t idx = (size_t)n * TT_LEN + t;
            zg[(size_t)b * NN * TT_LEN + idx] = acc[r] + xb[idx];
        }
    }
}

// ---------------- kernel D: conv pass 1 -> per-(row,h) partial sum / sumsq ----------------
__global__ void k_convstats(const float* __restrict__ zg, const float* __restrict__ cw,
                            const float* __restrict__ cb,
                            float* __restrict__ psum, float* __restrict__ psq) {
    __shared__ float row[1016];
    __shared__ float rs[256], rq[256];
    int tid  = threadIdx.x;
    int grow = blockIdx.x;                   // (b*N+n)
    const float* src = zg + (size_t)grow * TT_LEN;
    for (int i = tid; i < 1016; i += 256) {
        int s = i - 7;
        row[i] = (s >= 0 && s < TT_LEN) ? src[s] : 0.f;
    }
    __syncthreads();
    int h = tid >> 4, sub = tid & 15;
    float w[15];
    #pragma unroll
    for (int k = 0; k < 15; ++k) w[k] = cw[h * 15 + k];
    float bias = cb[h];
    float s = 0.f, q = 0.f;
    for (int t = sub; t < TT_LEN; t += 16) {
        float v = bias;
        #pragma unroll
        for (int k = 0; k < 15; ++k) v += w[k] * row[t + k];
        s += v; q += v * v;
    }
    rs[tid] = s; rq[tid] = q;
    __syncthreads();
    #pragma unroll
    for (int off = 8; off; off >>= 1) {
        if (sub < off) { rs[tid] += rs[tid + off]; rq[tid] += rq[tid + off]; }
        __syncthreads();
    }
    if (sub == 0) { psum[grow * HH + h] = rs[tid]; psq[grow * HH + h] = rq[tid]; }
}

// ---------------- kernel D2: finalize BN scale/shift per channel ----------------
__global__ void k_bnstats(const float* __restrict__ psum, const float* __restrict__ psq,
                          const float* __restrict__ gamma, const float* __restrict__ beta,
                          float* __restrict__ stats) {
    __shared__ float rs[256], rq[256];
    int h = blockIdx.x, tid = threadIdx.x;
    float s = 0.f, q = 0.f;
    for (int r = tid; r < BB * NN; r += 256) { s += psum[r * HH + h]; q += psq[r * HH + h]; }
    rs[tid] = s; rq[tid] = q;
    __syncthreads();
    #pragma unroll
    for (int off = 128; off; off >>= 1) {
        if (tid < off) { rs[tid] += rs[tid + off]; rq[tid] += rq[tid + off]; }
        __syncthreads();
    }
    if (tid == 0) {
        const float cnt = (float)BB * NN * TT_LEN;
        float mu  = rs[0] / cnt;
        float var = rq[0] / cnt - mu * mu;
        float a   = gamma[h] * rsqrtf(var + 1e-5f);
        stats[h]      = a;                    // scale
        stats[HH + h] = beta[h] - mu * a;     // shift
    }
}

// ---------------- kernel E: conv pass 2 + BN + ELU + LIF scan -> firing rate ----------------
#define ROWS_E 8
#define RL 1016
__global__ void k_snn(const float* __restrict__ zg, const float* __restrict__ cw,
                      const float* __restrict__ cb, const float* __restrict__ stats,
                      float* __restrict__ zfeat) {
    __shared__ float sm[ROWS_E * RL];
    int tid  = threadIdx.x;                  // 0..127 : 8 rows x 16 h
    int row0 = blockIdx.x * ROWS_E;
    for (int i = tid; i < ROWS_E * RL; i += 128) {
        int r = i / RL, c = i % RL;
        int s = c - 7;
        sm[i] = (s >= 0 && s < TT_LEN) ? zg[(size_t)(row0 + r) * TT_LEN + s] : 0.f;
    }
    __syncthreads();
    int r = tid >> 4, h = tid & 15;
    float w[15];
    #pragma unroll
    for (int k = 0; k < 15; ++k) w[k] = cw[h * 15 + k];
    float bias = cb[h];
    float a = stats[h], c = stats[HH + h];
    const float* base = sm + r * RL;
    float win[15];
    #pragma unroll
    for (int k = 0; k < 15; ++k) win[k] = base[k];
    float v = 0.f, cnt = 0.f;
    for (int t = 0; t < TT_LEN; ++t) {
        float z = bias;
        #pragma unroll
        for (int k = 0; k < 15; ++k) z += w[k] * win[k];
        z = z * a + c;                              // batchnorm affine
        z = (z > 0.f) ? z : (__expf(z) - 1.f);      // ELU
        v = v * 0.5f + z;                           // leak (1 - 1/TAU)
        float spk = (v - 0.5f) > 0.f ? 1.f : 0.f;   // spike(v - VTH)
        v -= spk * 0.5f;
        cnt += spk;
        #pragma unroll
        for (int k = 0; k < 14; ++k) win[k] = win[k + 1];
        win[14] = base[t + 15];                     // t=999 -> idx 1014 (zero pad)
    }
    int grow = row0 + r;
    int b = grow >> 6, n = grow & 63;
    zfeat[(size_t)b * (HH * NN) + h * NN + n] = cnt * (1.0f / TT_LEN);
}

// ---------------- kernel F: logits = z_feat @ fc_w.T + fc_b ----------------
__global__ void k_fc(const float* __restrict__ zfeat, const float* __restrict__ fcw,
                     const float* __restrict__ fcb, float* __restrict__ logits) {
    __shared__ float red[4][128];
    int b = blockIdx.x, tid = threadIdx.x;
    float acc[4] = {0.f, 0.f, 0.f, 0.f};
    for (int i = tid; i < HH * NN; i += 128) {
        float zv = zfeat[(size_t)b * HH * NN + i];
        #pragma unroll
        for (int o = 0; o < 4; ++o) acc[o] += zv * fcw[o * HH * NN + i];
    }
    #pragma unroll
    for (int o = 0; o < 4; ++o) red[o][tid] = acc[o];
    __syncthreads();
    #pragma unroll
    for (int off = 64; off; off >>= 1) {
        if (tid < off) {
            #pragma unroll
            for (int o = 0; o < 4; ++o) red[o][tid] += red[o][tid + off];
        }
        __syncthreads();
    }
    if (tid < 4) logits[b * 4 + tid] = red[tid][0] + fcb[tid];
}

// ---------------- launcher ----------------
extern "C" void kernel_launch(void* const* d_in, const int* in_sizes, int n_in,
                              void* d_out, int out_size, void* d_ws, size_t ws_size,
                              hipStream_t stream) {
    const float* x  = (const float*)d_in[0];
    const float* qw = (const float*)d_in[1];
    const float* qb = (const float*)d_in[2];
    const float* kw = (const float*)d_in[3];
    const float* kb = (const float*)d_in[4];
    const float* cw = (const float*)d_in[5];
    const float* cb = (const float*)d_in[6];
    const float* gm = (const float*)d_in[7];
    const float* bt = (const float*)d_in[8];
    const float* fw = (const float*)d_in[9];
    const float* fb = (const float*)d_in[10];

    // output layout: [logits(512) | z_feat(131072) | w(524288)]
    float* out    = (float*)d_out;
    float* logits = out;
    float* zfeat  = out + BB * 4;
    float* wmat   = out + BB * 4 + (size_t)BB * HH * NN;

    // workspace layout (~33.9 MB of floats)
    float* ws     = (float*)d_ws;
    float* zg     = ws;                              // B*N*T = 8,192,000
    float* energy = zg + (size_t)BB * NN * TT_LEN;   // 8192
    float* psum   = energy + BB * NN;                // B*N*H = 131072
    float* psq    = psum + (size_t)BB * NN * HH;     // 131072
    float* stats  = psq  + (size_t)BB * NN * HH;     // 32

    k_energy   <<<BB * NN / 8, 256, 0, stream>>>(x, energy);
    k_attn     <<<BB, NN, 0, stream>>>(energy, qw, qb, kw, kb, wmat);
    {
        int tiles = BB * (NN / 16) * ((TT_LEN + 15) / 16);  // 32256
        k_gcn  <<<tiles / 8, 256, 0, stream>>>(wmat, x, zg);
    }
    k_convstats<<<BB * NN, 256, 0, stream>>>(zg, cw, cb, psum, psq);
    k_bnstats  <<<HH, 256, 0, stream>>>(psum, psq, gm, bt, stats);
    k_snn      <<<BB * NN / ROWS_E, 128, 0, stream>>>(zg, cw, cb, stats, zfeat);
    k_fc       <<<BB, 128, 0, stream>>>(zfeat, fw, fb, logits);
}